// LocalPatternFilter_32418413150217
// MI455X (gfx1250) — compile-verified
//
#include <hip/hip_runtime.h>
#include <math.h>

#define IN_CH   4
#define OUT_CH  128
#define KWIN    1023
#define BATCH   32
#define LENGTH  131072
#define NWIN    (BATCH * IN_CH * OUT_CH)   // 16384 windows
#define TWOK    2046
#define RBIAS   16                          // lag-accumulator bias (sinks tile-0 negative-lag dups)

typedef float v2f __attribute__((ext_vector_type(2)));
typedef float v8f __attribute__((ext_vector_type(8)));

// ---------------------------------------------------------------------------
// Kernel 1: mx[p] = max over (b,c,n) of |X[b,c,1024*p + n]|, n in [0,1023)
// ---------------------------------------------------------------------------
__global__ __launch_bounds__(256)
void lpf_max_kernel(const float* __restrict__ X, float* __restrict__ mx) {
    const int p   = blockIdx.x;            // 0..127
    const int tid = threadIdx.x;
    const int start = p << 10;             // 1024 * p
    float m = 0.0f;
    for (int bc = 0; bc < BATCH * IN_CH; ++bc) {
        const float* row = X + (size_t)bc * LENGTH + start;
        for (int n = tid; n < KWIN; n += 256)
            m = fmaxf(m, fabsf(row[n]));
    }
    __shared__ float red[256];
    red[tid] = m;
    __syncthreads();
    for (int s = 128; s > 0; s >>= 1) {
        if (tid < s) red[tid] = fmaxf(red[tid], red[tid + s]);
        __syncthreads();
    }
    if (tid == 0) mx[p] = fmaxf(red[0], 0.0f);
}

// ---------------------------------------------------------------------------
// Kernel 2: per-window Hann-scaled autocorrelation via V_WMMA_F32_16X16X4_F32,
// then mirrored/ifftshifted writeout. One wave32 per window, 4 waves/block.
// ---------------------------------------------------------------------------
__global__ __launch_bounds__(128)
void lpf_acov_kernel(const float* __restrict__ X,
                     const float* __restrict__ mx,
                     float* __restrict__ out) {
    __shared__ float hann[1024];
    __shared__ float wbuf[4][2048];    // zero-padded scaled window, per wave
    __shared__ float rbuf[4][1056];    // biased lag accumulator, per wave

    const int tid = threadIdx.x;

    // Periodic Hann window (shared by all 4 waves in the block)
    for (int n = tid; n < 1024; n += 128)
        hann[n] = 0.5f - 0.5f * cosf((float)(6.283185307179586 / 1023.0) * (float)n);
    __syncthreads();

    const int wave = tid >> 5;
    const int lane = tid & 31;
    const int hs   = lane >> 4;        // half-wave select (0/1)
    const int la   = lane & 15;

    const int wid = blockIdx.x * 4 + wave;   // 0..16383
    const int p   = wid & 127;
    const int bc  = wid >> 7;                // 0..127
    const int b   = bc >> 2;
    const int c   = bc & 3;

    float* w = wbuf[wave];
    float* r = rbuf[wave];

    const float invm = 1.0f / mx[p];
    const float* xw  = X + (size_t)bc * LENGTH + ((size_t)p << 10);

    // Phase A: load + Hann*1/mx scale + zero-pad to 2048; zero lag accumulator
    for (int n = lane; n < 2048; n += 32) {
        float v = 0.0f;
        if (n < KWIN) v = xw[n] * hann[n] * invm;
        w[n] = v;
    }
    for (int n = lane; n < 1056; n += 32) r[n] = 0.0f;

    // Phase B: hoist the 16 B fragments (whole window) into registers.
    // f32 4x16 B layout: VGPR0 -> K = 2*hs, VGPR1 -> K = 2*hs+1, N = la
    // => B[k][j] = w[n0 + 16k + j] is lane la reading w[n0 + 32*hs + la], +16.
    v2f Bf[16];
#pragma unroll
    for (int cc = 0; cc < 16; ++cc) {
        const int base = (cc << 6) + (hs << 5) + la;
        Bf[cc].x = w[base];
        Bf[cc].y = w[base + 16];
    }

    // Phase C: 65 lag tiles; accumulate 16 chunks of n per tile in the C
    // operand, then scatter diagonals (lag = t_base + i - j) into r[].
    // Biased index RBIAS+lag is always >= 1, so the scatter is branch-free;
    // tile-0 negative-lag duplicates fall into padding slots r[1..15].
    const int sbase = RBIAS + (hs << 3) - la;   // per-lane constant part
    for (int T = 0; T < 65; ++T) {
        const int tb = T << 4;
        v8f acc = {0.0f, 0.0f, 0.0f, 0.0f, 0.0f, 0.0f, 0.0f, 0.0f};
#pragma unroll
        for (int cc = 0; cc < 16; ++cc) {
            const int ab = tb + (cc << 6) + (hs << 5) + la;
            v2f Af;
            Af.x = w[ab];          // A[i][k] = w[tb + n0 + 16k + i]
            Af.y = w[ab + 16];
            acc = __builtin_amdgcn_wmma_f32_16x16x4_f32(
                false, Af, false, Bf[cc], (short)0, acc, false, false);
        }
        // D layout: lane, VGPR v -> (M = v + 8*hs, N = la); lag = tb + M - N.
        float* rt = r + (tb + sbase);
#pragma unroll
        for (int v = 0; v < 8; ++v) {
            atomicAdd(&rt[v], acc[v]);
        }
    }

    // Phase D: writeout. out[(b+16)%32, p, (c+2)%4, t] = r[|t-1023|].
    // t = 0 maps to lag 1023, which is exactly 0 by zero-padding, matching
    // the reference's acov[1023] slot -> fully branch-free gather + store.
    const int b_out = (b + 16) & 31;
    const int c_out = (c + 2) & 3;
    float* op = out + ((((size_t)b_out * OUT_CH) + p) * IN_CH + c_out) * (size_t)TWOK;
    const float* rr = r + RBIAS;
    for (int t = lane; t < TWOK; t += 32) {
        const int d = t - 1023;
        op[t] = rr[d < 0 ? -d : d];
    }
}

// ---------------------------------------------------------------------------
extern "C" void kernel_launch(void* const* d_in, const int* in_sizes, int n_in,
                              void* d_out, int out_size, void* d_ws, size_t ws_size,
                              hipStream_t stream) {
    (void)in_sizes; (void)n_in; (void)out_size; (void)ws_size;
    const float* X  = (const float*)d_in[0];
    float*       mx = (float*)d_ws;           // 128 floats of scratch
    float*       out = (float*)d_out;

    lpf_max_kernel<<<OUT_CH, 256, 0, stream>>>(X, mx);
    lpf_acov_kernel<<<NWIN / 4, 128, 0, stream>>>(X, mx, out);
}